// TripletSemihardLoss_46540265619748
// MI455X (gfx1250) — compile-verified
//
#include <hip/hip_runtime.h>
#include <hip/hip_bf16.h>
#include <math.h>

#define N 384
#define D 128
#define TRIPLET_MARGIN 0.2f

typedef __attribute__((ext_vector_type(2))) float v2f;
typedef __attribute__((ext_vector_type(8))) float v8f;

// ---------------------------------------------------------------------------
// Kernel 1: row L2-normalize, and sq[i] = sum(xn[i]^2) (matches reference,
// which recomputes the squared norm of the normalized rows).
// One block (128 threads) per row.
// ---------------------------------------------------------------------------
__global__ __launch_bounds__(128)
void tsh_normalize(const float* __restrict__ x, float* __restrict__ xn,
                   float* __restrict__ sq) {
    __shared__ float red[128];
    const int row = blockIdx.x;
    const int t   = threadIdx.x;
    float v = x[row * D + t];
    red[t] = v * v;
    __syncthreads();
    for (int s = 64; s > 0; s >>= 1) {
        if (t < s) red[t] += red[t + s];
        __syncthreads();
    }
    const float inv = 1.0f / sqrtf(red[0]);
    __syncthreads();
    const float nv = v * inv;
    xn[row * D + t] = nv;
    red[t] = nv * nv;
    __syncthreads();
    for (int s = 64; s > 0; s >>= 1) {
        if (t < s) red[t] += red[t + s];
        __syncthreads();
    }
    if (t == 0) sq[row] = red[0];
}

// ---------------------------------------------------------------------------
// Kernel 2: Gram matrix G = xn @ xn^T via V_WMMA_F32_16X16X4_F32 (exact f32),
// and dist = clamp(sq_i + sq_j - 2G, 1e-12).
// One wave (32 threads) per 16x16 output tile; 24x24 = 576 tiles.
// f32 A (16x4) fragment: lanes 0-15 hold K=0,1; lanes 16-31 hold K=2,3.
// B (4x16) is the symmetric layout, and since B[k][n] = xn[j0+n][k], both
// fragments are a contiguous float2 load from one row of xn.
// ---------------------------------------------------------------------------
__global__ __launch_bounds__(32)
void tsh_gram_wmma(const float* __restrict__ xn, const float* __restrict__ sq,
                   float* __restrict__ G, float* __restrict__ dist) {
    const int lane = threadIdx.x;            // 0..31, full wave, EXEC all-1s
    const int tile = blockIdx.x;             // 0..575
    const int i0 = (tile % (N / 16)) * 16;
    const int j0 = (tile / (N / 16)) * 16;
    const int r15 = lane & 15;
    const int kk  = (lane >> 4) << 1;        // 0 for lanes 0-15, 2 for 16-31

    const float* arow = xn + (size_t)(i0 + r15) * D + kk;
    const float* brow = xn + (size_t)(j0 + r15) * D + kk;

    v8f acc = {};
#pragma unroll 8
    for (int k0 = 0; k0 < D; k0 += 4) {
        v2f a = *(const v2f*)(arow + k0);
        v2f b = *(const v2f*)(brow + k0);
        // (neg_a, A, neg_b, B, c_mod, C, reuse_a, reuse_b)
        acc = __builtin_amdgcn_wmma_f32_16x16x4_f32(
            false, a, false, b, (short)0, acc, false, false);
    }

    // C/D layout: VGPR r -> lanes 0-15: (M=r, N=lane); lanes 16-31: (M=8+r).
    const int col   = j0 + r15;
    const int rbase = i0 + ((lane >> 4) << 3);
    const float sqc = sq[col];
#pragma unroll
    for (int r = 0; r < 8; ++r) {
        const int row = rbase + r;
        const float g = acc[r];
        G[row * N + col] = g;
        const float dd = sq[row] + sqc - 2.0f * g;
        dist[row * N + col] = fmaxf(dd, 1e-12f);
    }
}

// ---------------------------------------------------------------------------
// Kernel 3: per-anchor semi-hard mining + partial sums.
// One block (256 threads) per anchor x. Distance row + labels live in LDS.
// Only positive pairs need the O(N) inner negative scan.
// Partials (deterministic per block): [loss, pos_cnt, ap_sum, an_sum, an_cnt]
// ---------------------------------------------------------------------------
__global__ __launch_bounds__(256)
void tsh_mine(const float* __restrict__ G, const float* __restrict__ dist,
              const int* __restrict__ labels, float* __restrict__ partials) {
    __shared__ float drow[N];
    __shared__ int   lab[N];
    __shared__ float red[256];

    const int x = blockIdx.x;
    const int t = threadIdx.x;

    for (int k = t; k < N; k += 256) {
        drow[k] = dist[x * N + k];
        lab[k]  = labels[k];
    }
    __syncthreads();
    const int mylab = lab[x];

    // Row min and max-over-negatives (for negatives_inside fallback).
    float lmin = 1e30f, lnegmax = -1e30f;
    for (int k = t; k < N; k += 256) {
        const float d = drow[k];
        lmin = fminf(lmin, d);
        if (lab[k] != mylab) lnegmax = fmaxf(lnegmax, d);
    }
    red[t] = lmin; __syncthreads();
    for (int s = 128; s > 0; s >>= 1) {
        if (t < s) red[t] = fminf(red[t], red[t + s]);
        __syncthreads();
    }
    const float rmin = red[0];
    __syncthreads();
    red[t] = lnegmax; __syncthreads();
    for (int s = 128; s > 0; s >>= 1) {
        if (t < s) red[t] = fmaxf(red[t], red[t + s]);
        __syncthreads();
    }
    const float negmax = red[0];
    __syncthreads();
    // neg_in = rmin + max over negatives of (d - rmin); = rmin if no negatives
    const float neg_in = (negmax > -1e29f) ? fmaxf(negmax, rmin) : rmin;

    float loss = 0.0f, pc = 0.0f, ap = 0.0f, an = 0.0f, ac = 0.0f;
    for (int y = t; y < N; y += 256) {
        const bool same = (lab[y] == mylab);
        const float cosv = G[x * N + y];
        if (same) {
            if (y != x) {
                pc += 1.0f;
                ap += cosv;
                const float tv = drow[y];
                float m = 1e30f;
                for (int k = 0; k < N; ++k) {
                    const float dk = drow[k];
                    const bool ok = (lab[k] != mylab) && (dk > tv);
                    m = ok ? fminf(m, dk) : m;
                }
                const float semi = (m < 1e29f) ? m : neg_in;
                loss += fmaxf(0.0f, TRIPLET_MARGIN + tv - semi);
            }
        } else {
            ac += 1.0f;
            an += cosv;
        }
    }

    float vals[5] = {loss, pc, ap, an, ac};
#pragma unroll
    for (int i = 0; i < 5; ++i) {
        red[t] = vals[i];
        __syncthreads();
        for (int s = 128; s > 0; s >>= 1) {
            if (t < s) red[t] += red[t + s];
            __syncthreads();
        }
        if (t == 0) partials[x * 5 + i] = red[0];
        __syncthreads();
    }
}

// ---------------------------------------------------------------------------
// Kernel 4: deterministic final reduction over 384 anchor partials.
// ---------------------------------------------------------------------------
__global__ void tsh_finalize(const float* __restrict__ partials,
                             float* __restrict__ out) {
    if (threadIdx.x == 0 && blockIdx.x == 0) {
        float loss = 0.0f, pc = 0.0f, ap = 0.0f, an = 0.0f, ac = 0.0f;
        for (int x = 0; x < N; ++x) {
            loss += partials[x * 5 + 0];
            pc   += partials[x * 5 + 1];
            ap   += partials[x * 5 + 2];
            an   += partials[x * 5 + 3];
            ac   += partials[x * 5 + 4];
        }
        out[0] = loss / pc;
        out[1] = ap / pc;
        out[2] = an / ac;
    }
}

extern "C" void kernel_launch(void* const* d_in, const int* in_sizes, int n_in,
                              void* d_out, int out_size, void* d_ws, size_t ws_size,
                              hipStream_t stream) {
    (void)in_sizes; (void)n_in; (void)out_size; (void)ws_size;
    const float* x      = (const float*)d_in[0];
    const int*   labels = (const int*)d_in[1];
    float*       out    = (float*)d_out;

    // Workspace layout (floats): xn[384*128] | sq[384] | G[384*384] |
    //                            dist[384*384] | partials[384*5]
    float* xn       = (float*)d_ws;
    float* sq       = xn + N * D;
    float* G        = sq + N;
    float* dist     = G + N * N;
    float* partials = dist + N * N;

    tsh_normalize<<<N, 128, 0, stream>>>(x, xn, sq);
    tsh_gram_wmma<<<(N / 16) * (N / 16), 32, 0, stream>>>(xn, sq, G, dist);
    tsh_mine<<<N, 256, 0, stream>>>(G, dist, labels, partials);
    tsh_finalize<<<1, 1, 0, stream>>>(partials, out);
}